// LightGCN_37890201485521
// MI455X (gfx1250) — compile-verified
//
#include <hip/hip_runtime.h>
#include <hip/hip_bf16.h>
#include <stdint.h>

#define NUM_USERS 100000
#define NUM_ITEMS 50000
#define N_NODES   150000
#define DIM       64
#define NNZ_E     2400000
#define NQ        4096
#define HOPS      3

// ---------------------------------------------------------------------------
// gfx1250 async global->LDS via inline asm (unconditional, verifiable)
// ---------------------------------------------------------------------------
__device__ __forceinline__ uint32_t lds_off_of(const void* p) {
  // generic -> LDS(AS3) addrspacecast, then take the 32-bit LDS byte offset
  return (uint32_t)(uintptr_t)(__attribute__((address_space(3))) const void*)p;
}

__device__ __forceinline__ void async_gl2lds_b64(const void* gsrc, uint32_t ldsoff) {
  // GLOBAL_LOAD_ASYNC_TO_LDS_B64: vdst = 32-bit LDS byte address, vaddr = 64-bit
  asm volatile("global_load_async_to_lds_b64 %0, %1, off"
               :: "v"(ldsoff), "v"(gsrc)
               : "memory");
}

__device__ __forceinline__ void wait_async0() {
  asm volatile("s_wait_asynccnt 0" ::: "memory");
}

// ---------------------------------------------------------------------------
// 1) concat embeddings -> x0 and acc (float4 vectorized; split is f4-aligned)
// ---------------------------------------------------------------------------
__global__ void init_kernel(const float4* __restrict__ u4, const float4* __restrict__ it4,
                            float4* __restrict__ x4, float4* __restrict__ acc4) {
  int i = blockIdx.x * blockDim.x + threadIdx.x;
  const int total4 = N_NODES * (DIM / 4);          // 2.4M
  if (i >= total4) return;
  const int ub4 = NUM_USERS * (DIM / 4);           // 1.6M
  const float4 v = (i < ub4) ? u4[i] : it4[i - ub4];
  x4[i] = v;
  acc4[i] = v;
}

// ---------------------------------------------------------------------------
// 2) CSR build: histogram, single-block scan, cursor scatter (int2 {col,val})
// ---------------------------------------------------------------------------
__global__ void zero_int_kernel(int* __restrict__ p, int n) {
  int i = blockIdx.x * blockDim.x + threadIdx.x;
  if (i < n) p[i] = 0;
}

__global__ void count_kernel(const int* __restrict__ rows, int* __restrict__ counts) {
  int i = blockIdx.x * blockDim.x + threadIdx.x;
  if (i < NNZ_E) atomicAdd(&counts[rows[i]], 1);
}

__global__ void scan_kernel(const int* __restrict__ counts, int* __restrict__ row_ptr) {
  __shared__ int smem[1024];
  __shared__ int carry_s;
  if (threadIdx.x == 0) carry_s = 0;
  __syncthreads();
  for (int base = 0; base < N_NODES; base += 1024) {
    const int i = base + (int)threadIdx.x;
    const int v = (i < N_NODES) ? counts[i] : 0;
    smem[threadIdx.x] = v;
    __syncthreads();
    for (int off = 1; off < 1024; off <<= 1) {
      int t = (threadIdx.x >= (unsigned)off) ? smem[threadIdx.x - off] : 0;
      __syncthreads();
      smem[threadIdx.x] += t;
      __syncthreads();
    }
    const int incl = smem[threadIdx.x];
    const int c = carry_s;
    if (i < N_NODES) row_ptr[i] = c + incl - v;      // exclusive prefix
    const int chunk_total = smem[1023];
    __syncthreads();
    if (threadIdx.x == 0) carry_s = c + chunk_total;
    __syncthreads();
  }
  if (threadIdx.x == 0) row_ptr[N_NODES] = carry_s;  // == NNZ_E
}

__global__ void copy_int_kernel(const int* __restrict__ src, int* __restrict__ dst, int n) {
  int i = blockIdx.x * blockDim.x + threadIdx.x;
  if (i < n) dst[i] = src[i];
}

__global__ void scatter_kernel(const int* __restrict__ rows, const int* __restrict__ cols,
                               const float* __restrict__ vals, int* __restrict__ cursor,
                               int2* __restrict__ csr_cv) {
  int i = blockIdx.x * blockDim.x + threadIdx.x;
  if (i >= NNZ_E) return;
  const int r = rows[i];
  const int pos = atomicAdd(&cursor[r], 1);
  int2 cv;
  cv.x = cols[i];
  cv.y = __float_as_int(vals[i]);
  csr_cv[pos] = cv;                                  // 8B interleaved CSR record
}

// ---------------------------------------------------------------------------
// 3) SpMM: one wave32 per row; lane owns float2 of the 64-dim row.
//    Edge {col,val} chunks (32 at a time, 8B each) staged to LDS with
//    global_load_async_to_lds_b64; no cross-wave barriers needed.
// ---------------------------------------------------------------------------
__global__ void __launch_bounds__(256)
spmm_kernel(const int* __restrict__ row_ptr, const int2* __restrict__ csr_cv,
            const float* __restrict__ x, float* __restrict__ y,
            float* __restrict__ acc) {
  __shared__ int2 s_cv[8][32];                       // 2 KB
  const int w    = threadIdx.x >> 5;                 // wave id = local row
  const int lane = threadIdx.x & 31;
  const int row  = blockIdx.x * 8 + w;
  if (row >= N_NODES) return;

  const int start = row_ptr[row];
  const int end   = row_ptr[row + 1];
  const int d2    = lane * 2;
  const uint32_t my_lds = lds_off_of(&s_cv[w][lane]);

  float sx = 0.f, sy = 0.f;
  for (int base = start; base < end; base += 32) {
    const int cnt = min(32, end - base);
    if (lane < cnt) {
      async_gl2lds_b64(&csr_cv[base + lane], my_lds);  // {col,val} -> LDS
    }
    if (base + 32 < end && lane == 0) {
      __builtin_prefetch(&csr_cv[base + 32], 0, 0);    // global_prefetch_b8
    }
    wait_async0();                                     // s_wait_asynccnt 0
    #pragma unroll 4
    for (int k = 0; k < cnt; ++k) {
      const int2  cv = s_cv[w][k];                     // ds_load_b64, broadcast
      const float v  = __int_as_float(cv.y);
      const float2 xr = *(const float2*)(x + ((size_t)cv.x * DIM + d2));
      sx = fmaf(v, xr.x, sx);
      sy = fmaf(v, xr.y, sy);
    }
  }

  const size_t o = (size_t)row * DIM + d2;
  float2 r; r.x = sx; r.y = sy;
  *(float2*)(y + o) = r;                               // x for next hop
  float2 a = *(const float2*)(acc + o);
  a.x += sx; a.y += sy;
  *(float2*)(acc + o) = a;                             // running hop sum
}

// ---------------------------------------------------------------------------
// 4) final gather of 4096 users + 4096 items, scaled 1/(HOP+1) (float4)
// ---------------------------------------------------------------------------
__global__ void gather_kernel(const float4* __restrict__ acc4, const int* __restrict__ uids,
                              const int* __restrict__ iids, float4* __restrict__ out4) {
  int t = blockIdx.x * blockDim.x + threadIdx.x;
  const int total4 = 2 * NQ * (DIM / 4);               // 131072
  if (t >= total4) return;
  const int i  = t >> 4;                               // output row
  const int dv = t & 15;                               // float4 within row
  const int src = (i < NQ) ? uids[i] : (NUM_USERS + iids[i - NQ]);
  float4 v = acc4[(size_t)src * (DIM / 4) + dv];
  v.x *= 0.25f; v.y *= 0.25f; v.z *= 0.25f; v.w *= 0.25f;
  out4[t] = v;
}

// ---------------------------------------------------------------------------
extern "C" void kernel_launch(void* const* d_in, const int* in_sizes, int n_in,
                              void* d_out, int out_size, void* d_ws, size_t ws_size,
                              hipStream_t stream) {
  const float* user_emb  = (const float*)d_in[0];
  const float* item_emb  = (const float*)d_in[1];
  const float* edge_vals = (const float*)d_in[2];
  const int*   edge_rows = (const int*)d_in[3];
  const int*   edge_cols = (const int*)d_in[4];
  const int*   user_ids  = (const int*)d_in[5];
  const int*   item_ids  = (const int*)d_in[6];

  const size_t nd = (size_t)N_NODES * DIM;
  float* acc     = (float*)d_ws;                 // 38.4 MB
  float* x0      = acc + nd;                     // 38.4 MB
  float* x1      = x0 + nd;                      // 38.4 MB
  int*   row_ptr = (int*)(x1 + nd);              // N_NODES+1
  int*   counts  = row_ptr + (N_NODES + 1);      // N_NODES
  int*   cursor  = counts + N_NODES;             // N_NODES
  int2*  csr_cv  = (int2*)(cursor + N_NODES);    // NNZ_E * 8B = 19.2 MB

  const int T = 256;
  init_kernel<<<(N_NODES * (DIM / 4) + T - 1) / T, T, 0, stream>>>(
      (const float4*)user_emb, (const float4*)item_emb, (float4*)x0, (float4*)acc);
  zero_int_kernel<<<(N_NODES + T - 1) / T, T, 0, stream>>>(counts, N_NODES);
  count_kernel<<<(NNZ_E + T - 1) / T, T, 0, stream>>>(edge_rows, counts);
  scan_kernel<<<1, 1024, 0, stream>>>(counts, row_ptr);
  copy_int_kernel<<<(N_NODES + T - 1) / T, T, 0, stream>>>(row_ptr, cursor, N_NODES);
  scatter_kernel<<<(NNZ_E + T - 1) / T, T, 0, stream>>>(edge_rows, edge_cols, edge_vals,
                                                        cursor, csr_cv);

  float* xin = x0; float* xout = x1;
  for (int h = 0; h < HOPS; ++h) {
    spmm_kernel<<<(N_NODES + 7) / 8, T, 0, stream>>>(row_ptr, csr_cv, xin, xout, acc);
    float* t = xin; xin = xout; xout = t;
  }

  gather_kernel<<<(2 * NQ * (DIM / 4) + T - 1) / T, T, 0, stream>>>(
      (const float4*)acc, user_ids, item_ids, (float4*)d_out);
}